// bce_loss_82222853914699
// MI455X (gfx1250) — compile-verified
//
#include <hip/hip_runtime.h>
#include <hip/hip_bf16.h>

typedef __attribute__((ext_vector_type(2))) float v2f;
typedef __attribute__((ext_vector_type(8))) float v8f;

#define ANUM 4096            // anchors
#define NTOT 8192            // ANUM * 2 views
#define WPB  4               // waves per block
#define COLS_PER_WAVE (NTOT / WPB)          // 2048
#define TILES_PER_WAVE (COLS_PER_WAVE / 16) // 128

static __device__ __forceinline__ float redmax16(float x) {
#pragma unroll
  for (int m = 8; m >= 1; m >>= 1) x = fmaxf(x, __shfl_xor(x, m, 32));
  return x;
}
static __device__ __forceinline__ float redsum16(float x) {
#pragma unroll
  for (int m = 8; m >= 1; m >>= 1) x += __shfl_xor(x, m, 32);
  return x;
}

// 16x16 tile of cf · cf^T via 16 chained V_WMMA_F32_16X16X4_F32 (K=64).
// afrag: per-lane A fragment (rows r0..r0+15), col selects this lane's B column.
static __device__ __forceinline__ v8f tile_gemm(const float* __restrict__ feats,
                                                int col, int koff,
                                                const v2f* __restrict__ afrag) {
  const float* bbase = feats + (size_t)(col & (ANUM - 1)) * 128 + (col >> 12) * 64;
  v8f c = {0.f, 0.f, 0.f, 0.f, 0.f, 0.f, 0.f, 0.f};
#pragma unroll
  for (int ks = 0; ks < 16; ++ks) {
    v2f b = *(const v2f*)(bbase + 4 * ks + koff);
    c = __builtin_amdgcn_wmma_f32_16x16x4_f32(
        /*neg_a=*/false, afrag[ks], /*neg_b=*/false, b,
        /*c_mod=*/(short)0, c, /*reuse_a=*/false, /*reuse_b=*/false);
  }
  return c;
}

__global__ void __launch_bounds__(WPB * 32)
supcon_fused(const float* __restrict__ feats, const int* __restrict__ labels,
             float* __restrict__ out) {
  __shared__ float sA[WPB][16];
  __shared__ float sB[WPB][16];

  const int tid  = threadIdx.x;
  const int wave = tid >> 5;
  const int lane = tid & 31;
  const int half = lane >> 4;   // C layout: rows m = 8*half + v
  const int l16  = lane & 15;   // C layout: column n = l16
  const int r0   = blockIdx.x * 16;
  const int c0w  = wave * COLS_PER_WAVE;
  const float tinv = 1.0f / 0.07f;

  // ---- preload A fragments (reused by all 3 sweeps) ----
  const int arow = r0 + l16;
  const float* abase =
      feats + (size_t)(arow & (ANUM - 1)) * 128 + (arow >> 12) * 64;
  v2f afrag[16];
#pragma unroll
  for (int ks = 0; ks < 16; ++ks)
    afrag[ks] = *(const v2f*)(abase + 4 * ks + 2 * half);

  int rlab[8];
#pragma unroll
  for (int v = 0; v < 8; ++v)
    rlab[v] = labels[(r0 + half * 8 + v) & (ANUM - 1)];

  // ============ Sweep 1: row max over all columns ============
  float lmax[8];
#pragma unroll
  for (int v = 0; v < 8; ++v) lmax[v] = -3.4e38f;
  for (int t = 0; t < TILES_PER_WAVE; ++t) {
    int col = c0w + t * 16 + l16;
    if (t + 1 < TILES_PER_WAVE) {
      int nc = col + 16;
      __builtin_prefetch(feats + (size_t)(nc & (ANUM - 1)) * 128 + (nc >> 12) * 64, 0, 0);
    }
    v8f c = tile_gemm(feats, col, 2 * half, afrag);
#pragma unroll
    for (int v = 0; v < 8; ++v) lmax[v] = fmaxf(lmax[v], c[v] * tinv);
  }
#pragma unroll
  for (int v = 0; v < 8; ++v) lmax[v] = redmax16(lmax[v]);
  if (l16 == 0) {
#pragma unroll
    for (int v = 0; v < 8; ++v) sA[wave][half * 8 + v] = lmax[v];
  }
  __syncthreads();
  float M[8];
#pragma unroll
  for (int v = 0; v < 8; ++v) {
    float m = sA[0][half * 8 + v];
#pragma unroll
    for (int w = 1; w < WPB; ++w) m = fmaxf(m, sA[w][half * 8 + v]);
    M[v] = m;
  }
  __syncthreads();

  // ============ Sweep 2: negatives exp-sum ============
  float ns[8];
#pragma unroll
  for (int v = 0; v < 8; ++v) ns[v] = 0.f;
  for (int t = 0; t < TILES_PER_WAVE; ++t) {
    int col = c0w + t * 16 + l16;
    int clab = labels[col & (ANUM - 1)];
    v8f c = tile_gemm(feats, col, 2 * half, afrag);
#pragma unroll
    for (int v = 0; v < 8; ++v) {
      float x = c[v] * tinv - M[v];
      ns[v] += (clab != rlab[v]) ? __expf(x) : 0.f;
    }
  }
#pragma unroll
  for (int v = 0; v < 8; ++v) ns[v] = redsum16(ns[v]);
  if (l16 == 0) {
#pragma unroll
    for (int v = 0; v < 8; ++v) sA[wave][half * 8 + v] = ns[v];
  }
  __syncthreads();
  float S[8];
#pragma unroll
  for (int v = 0; v < 8; ++v) {
    float s = 0.f;
#pragma unroll
    for (int w = 0; w < WPB; ++w) s += sA[w][half * 8 + v];
    S[v] = s;
  }
  __syncthreads();

  // ============ Sweep 3: positives log-prob sum ============
  float pacc[8], pcnt[8];
#pragma unroll
  for (int v = 0; v < 8; ++v) { pacc[v] = 0.f; pcnt[v] = 0.f; }
  for (int t = 0; t < TILES_PER_WAVE; ++t) {
    int col = c0w + t * 16 + l16;
    int clab = labels[col & (ANUM - 1)];
    v8f c = tile_gemm(feats, col, 2 * half, afrag);
#pragma unroll
    for (int v = 0; v < 8; ++v) {
      int i = r0 + half * 8 + v;                     // global row
      bool pos = (clab == rlab[v]) && (col != i);    // positive, diag excluded
      if (pos) {
        float x = c[v] * tinv - M[v];
        pacc[v] += x - __logf(__expf(x) + S[v]);
        pcnt[v] += 1.0f;
      }
    }
  }
#pragma unroll
  for (int v = 0; v < 8; ++v) {
    pacc[v] = redsum16(pacc[v]);
    pcnt[v] = redsum16(pcnt[v]);
  }
  if (l16 == 0) {
#pragma unroll
    for (int v = 0; v < 8; ++v) {
      sA[wave][half * 8 + v] = pacc[v];
      sB[wave][half * 8 + v] = pcnt[v];
    }
  }
  __syncthreads();

  if (tid < 16) {
    float pa = 0.f, pc = 0.f;
#pragma unroll
    for (int w = 0; w < WPB; ++w) { pa += sA[w][tid]; pc += sB[w][tid]; }
    float rowloss = -(pa / pc);          // TEMP/BASE_TEMP == 1
    rowloss = redsum16(rowloss);         // lanes 0..15 only, masks <= 8
    if (tid == 0) atomicAdd(out, rowloss * (1.0f / (float)NTOT));
  }
}

__global__ void zero_out(float* out) { out[0] = 0.f; }

extern "C" void kernel_launch(void* const* d_in, const int* in_sizes, int n_in,
                              void* d_out, int out_size, void* d_ws, size_t ws_size,
                              hipStream_t stream) {
  const float* feats  = (const float*)d_in[0];
  const int*   labels = (const int*)d_in[1];
  float*       out    = (float*)d_out;

  zero_out<<<1, 1, 0, stream>>>(out);
  supcon_fused<<<NTOT / 16, WPB * 32, 0, stream>>>(feats, labels, out);
}